// GINE_4879082848574
// MI455X (gfx1250) — compile-verified
//
#include <hip/hip_runtime.h>
#include <hip/hip_bf16.h>

#define D 64
#define WAVES_PER_BLOCK 8
#define BLOCK 256

typedef __attribute__((ext_vector_type(2))) float v2f;
typedef __attribute__((ext_vector_type(8))) float v8f;

// ---------------------------------------------------------------------------
// Kernel 1: h = (1 + eps) * x   (float4-vectorized streaming pass)
// ---------------------------------------------------------------------------
__global__ void gine_init_h(const float* __restrict__ x,
                            const float* __restrict__ eps,
                            float* __restrict__ h,
                            int n_vec4) {
  int i = blockIdx.x * BLOCK + threadIdx.x;
  if (i >= n_vec4) return;
  const float s = 1.0f + eps[0];
  float4 v = ((const float4*)x)[i];
  v.x *= s; v.y *= s; v.z *= s; v.w *= s;
  ((float4*)h)[i] = v;
}

// ---------------------------------------------------------------------------
// Kernel 2: per-edge  h[dst] += relu(x[src] + edge_attr)
// 16 lanes per edge, float4 per lane, f32 global atomics (L2-resident target)
// ---------------------------------------------------------------------------
__global__ void gine_edge_scatter(const float* __restrict__ x,
                                  const int* __restrict__ src,
                                  const int* __restrict__ dst,
                                  const float* __restrict__ ea,
                                  float* __restrict__ h,
                                  int n_edges) {
  const int tid = blockIdx.x * BLOCK + threadIdx.x;
  const int e = tid >> 4;
  if (e >= n_edges) return;
  const int c = (tid & 15) << 2;           // float4 slot within the 64-wide row
  const int s = src[e];
  const int d = dst[e];

  const float4 xv = *(const float4*)(x + (size_t)s * D + c);
  const float4 av = *(const float4*)(ea + (size_t)e * D + c);
  float4 m;
  m.x = fmaxf(xv.x + av.x, 0.0f);
  m.y = fmaxf(xv.y + av.y, 0.0f);
  m.z = fmaxf(xv.z + av.z, 0.0f);
  m.w = fmaxf(xv.w + av.w, 0.0f);

  float* p = h + (size_t)d * D + c;
  atomicAdd(p + 0, m.x);
  atomicAdd(p + 1, m.y);
  atomicAdd(p + 2, m.z);
  atomicAdd(p + 3, m.w);
}

// ---------------------------------------------------------------------------
// Kernel 3: out = relu(h @ W^T + b) via V_WMMA_F32_16X16X4_F32
// One wave computes a 16-row x 64-col output stripe (4 accumulator tiles).
//
// f32 16x4 A layout : lane L (M = L&15) holds K = {0,1} (L<16) or {2,3} (L>=16)
// f32 4x16 B layout : lane L (N = L&15) holds K = {0,1} (L<16) or {2,3} (L>=16)
// f32 16x16 C/D     : VGPR v, lane L -> M = v + ((L>=16)?8:0), N = L&15
// ---------------------------------------------------------------------------
__global__ void gine_gemm_relu(const float* __restrict__ h,
                               const float* __restrict__ W,     // [D_OUT, D_IN] row-major
                               const float* __restrict__ bias,  // [D_OUT]
                               float* __restrict__ out,
                               int n_nodes) {
  const int lane = threadIdx.x & 31;
  const int wave = blockIdx.x * WAVES_PER_BLOCK + (threadIdx.x >> 5);
  const int row0 = wave * 16;                  // 16-row output tile
  if (row0 >= n_nodes) return;                 // uniform per-wave exit: EXEC stays full

  const int m = lane & 15;                     // A-row / B-col within tile
  const int khalf = (lane >> 4) << 1;          // 0 for lanes 0-15, 2 for lanes 16-31
  int arow = row0 + m;
  if (arow >= n_nodes) arow = n_nodes - 1;     // clamp (N % 16 == 0 -> never taken)

  v8f acc0 = {}, acc1 = {}, acc2 = {}, acc3 = {};

#pragma unroll
  for (int k0 = 0; k0 < D; k0 += 4) {
    const v2f a  = *(const v2f*)(h + (size_t)arow * D + k0 + khalf);
    const v2f b0 = *(const v2f*)(W + (size_t)(0  + m) * D + k0 + khalf);
    const v2f b1 = *(const v2f*)(W + (size_t)(16 + m) * D + k0 + khalf);
    const v2f b2 = *(const v2f*)(W + (size_t)(32 + m) * D + k0 + khalf);
    const v2f b3 = *(const v2f*)(W + (size_t)(48 + m) * D + k0 + khalf);
    acc0 = __builtin_amdgcn_wmma_f32_16x16x4_f32(false, a, false, b0, (short)0, acc0, false, false);
    acc1 = __builtin_amdgcn_wmma_f32_16x16x4_f32(false, a, false, b1, (short)0, acc1, false, false);
    acc2 = __builtin_amdgcn_wmma_f32_16x16x4_f32(false, a, false, b2, (short)0, acc2, false, false);
    acc3 = __builtin_amdgcn_wmma_f32_16x16x4_f32(false, a, false, b3, (short)0, acc3, false, false);
  }

  const float bb0 = bias[0  + m];
  const float bb1 = bias[16 + m];
  const float bb2 = bias[32 + m];
  const float bb3 = bias[48 + m];
  const int mbase = row0 + ((lane >> 4) << 3);  // M = mbase + v

#pragma unroll
  for (int v = 0; v < 8; ++v) {
    const int r = mbase + v;
    if (r < n_nodes) {
      float* o = out + (size_t)r * D;
      o[0  + m] = fmaxf(acc0[v] + bb0, 0.0f);
      o[16 + m] = fmaxf(acc1[v] + bb1, 0.0f);
      o[32 + m] = fmaxf(acc2[v] + bb2, 0.0f);
      o[48 + m] = fmaxf(acc3[v] + bb3, 0.0f);
    }
  }
}

// ---------------------------------------------------------------------------
// Launch: init -> edge scatter (atomics) -> WMMA GEMM, all on `stream`
// Inputs: [0]=x [1]=edge_index(2,E) [2]=edge_attr [3]=W [4]=b [5]=eps
// ---------------------------------------------------------------------------
extern "C" void kernel_launch(void* const* d_in, const int* in_sizes, int n_in,
                              void* d_out, int out_size, void* d_ws, size_t ws_size,
                              hipStream_t stream) {
  const float* x  = (const float*)d_in[0];
  const int*   ei = (const int*)d_in[1];
  const float* ea = (const float*)d_in[2];
  const float* W  = (const float*)d_in[3];
  const float* b  = (const float*)d_in[4];
  const float* ep = (const float*)d_in[5];
  float* out = (float*)d_out;

  const int n_nodes = in_sizes[0] / D;
  const int n_edges = in_sizes[1] / 2;
  const int* src = ei;             // edge_index[0, :]
  const int* dst = ei + n_edges;   // edge_index[1, :]

  float* h = (float*)d_ws;         // [n_nodes, D] accumulation buffer

  // 1) h = (1+eps)*x
  {
    int n_vec4 = (n_nodes * D) / 4;
    int grid = (n_vec4 + BLOCK - 1) / BLOCK;
    gine_init_h<<<grid, BLOCK, 0, stream>>>(x, ep, h, n_vec4);
  }
  // 2) scatter-add relu(x[src] + edge_attr) into h[dst]
  {
    long long work = (long long)n_edges * 16;
    int grid = (int)((work + BLOCK - 1) / BLOCK);
    gine_edge_scatter<<<grid, BLOCK, 0, stream>>>(x, src, dst, ea, h, n_edges);
  }
  // 3) out = relu(h @ W^T + b) with f32 WMMA
  {
    int tiles = (n_nodes + 15) / 16;
    int grid = (tiles + WAVES_PER_BLOCK - 1) / WAVES_PER_BLOCK;
    gine_gemm_relu<<<grid, BLOCK, 0, stream>>>(h, W, b, out, n_nodes);
  }
}